// involution_15882789061234
// MI455X (gfx1250) — compile-verified
//
#include <hip/hip_runtime.h>
#include <hip/hip_bf16.h>
#include <stdint.h>

typedef __attribute__((ext_vector_type(2))) float v2f;
typedef __attribute__((ext_vector_type(8))) float v8f;

// Problem constants (from reference)
#define CCH   256   // channels
#define KW    7     // involution kernel
#define GC    16    // group channels
#define NG    16    // groups
#define RED   64    // reduction channels
#define HH    56
#define WW    56
#define KK2   49    // K*K
#define KOUT  784   // K*K*G

#define TW    14            // pixel strip width (56 = 4*14)
#define SW    (TW + KW - 1) // slab width = 20
#define XSTR  257           // padded LDS stride for x tile
#define TSTR  65            // padded LDS stride for t tile

// Async global->LDS copy (CDNA5 GLOBAL_LOAD_ASYNC_TO_LDS_B32, ASYNCcnt-tracked).
// saddr form: LDS byte-offset VGPR, 32-bit byte voffset VGPR, SGPR-pair base.
__device__ __forceinline__ void async_g2l_b32(unsigned lds_off, unsigned voff,
                                              const float* __restrict__ base) {
    asm volatile("global_load_async_to_lds_b32 %0, %1, %2"
                 :: "v"(lds_off), "v"(voff), "s"(base)
                 : "memory");
}

__device__ __forceinline__ void wait_async0() {
    asm volatile("s_wait_asynccnt 0x0" ::: "memory");
}

__global__ __launch_bounds__(128) void involution_fused_kernel(
    const float* __restrict__ x,  const float* __restrict__ w1,
    const float* __restrict__ b1, const float* __restrict__ w2,
    const float* __restrict__ b2, float* __restrict__ out)
{
    __shared__ float sm_t[16 * TSTR];        //  4160 B : t tile (16 x 64, padded)
    __shared__ float sm_wgt[16 * KOUT];      // 50176 B : weight tile; head reused as x tile
    __shared__ float sm_slab[GC * KW * SW];  //  8960 B : per-group x slab (16 x 7 x 20)

    const int tid  = threadIdx.x;            // 0..127
    const int lane = tid & 31;
    const int wave = tid >> 5;               // 0..3
    const int half = lane >> 4;              // 0 = lanes 0-15, 1 = lanes 16-31
    const int l16  = lane & 15;

    const int strip = blockIdx.x;            // 0..895
    const int w0 = (strip & 3) * TW;
    const int bh = strip >> 2;
    const int h  = bh % HH;
    const int b  = bh / HH;

    float* xt = sm_wgt;                      // x tile [16][XSTR], dead after stage 1

    // ---- stage 0: async-stage x tile (16 pixels x 256 channels) into LDS ----
    {
        const unsigned base_e = (unsigned)((b * CCH) * (HH * WW) + h * WW + w0);
        for (int i = tid; i < 16 * CCH; i += 128) {
            const int c = i >> 4;
            const int m = i & 15;
            const int mm = (m < TW) ? m : (TW - 1);   // pad rows 14,15 (never stored)
            const unsigned voff = 4u * (base_e + (unsigned)(c * (HH * WW) + mm));
            const unsigned loff = (unsigned)(uintptr_t)(&xt[m * XSTR + c]);
            async_g2l_b32(loff, voff, x);
        }
        wait_async0();
    }
    __syncthreads();

    // ---- stage 1: t = relu(x_tile[16x256] * w1^T[256x64] + b1) ----
    // wave w computes output columns [16w, 16w+16); dual accumulators for XDL ILP
    {
        const int o0 = wave * 16;
        v8f acc0 = {0.f, 0.f, 0.f, 0.f, 0.f, 0.f, 0.f, 0.f};
        v8f acc1 = {0.f, 0.f, 0.f, 0.f, 0.f, 0.f, 0.f, 0.f};
        const int kb_off = half * 2;
        #pragma unroll 2
        for (int k0 = 0; k0 < CCH; k0 += 8) {
            {
                const int kb = k0 + kb_off;
                v2f a, bb;
                a.x = xt[l16 * XSTR + kb];
                a.y = xt[l16 * XSTR + kb + 1];
                const float* wp = w1 + (o0 + l16) * CCH + kb;
                bb.x = wp[0];
                bb.y = wp[1];
                acc0 = __builtin_amdgcn_wmma_f32_16x16x4_f32(
                    false, a, false, bb, (short)0, acc0, false, false);
            }
            {
                const int kb = k0 + 4 + kb_off;
                v2f a, bb;
                a.x = xt[l16 * XSTR + kb];
                a.y = xt[l16 * XSTR + kb + 1];
                const float* wp = w1 + (o0 + l16) * CCH + kb;
                bb.x = wp[0];
                bb.y = wp[1];
                acc1 = __builtin_amdgcn_wmma_f32_16x16x4_f32(
                    false, a, false, bb, (short)0, acc1, false, false);
            }
        }
        const float bias = b1[o0 + l16];
        #pragma unroll
        for (int r = 0; r < 8; ++r) {
            const int m = r + half * 8;                 // C/D layout: vgpr r, half -> row
            float v = acc0[r] + acc1[r] + bias;
            sm_t[m * TSTR + o0 + l16] = (v > 0.f) ? v : 0.f;
        }
    }
    __syncthreads();   // t complete; xt region free for weight tile

    // ---- stage 2: weight = t[16x64] * w2^T[64x784] + b2 ----
    // 49 column-chunks of 16, round-robin over waves (wave-uniform => EXEC all ones)
    for (int j = wave; j < KK2; j += 4) {
        const int col0 = j * 16;
        v8f acc0 = {0.f, 0.f, 0.f, 0.f, 0.f, 0.f, 0.f, 0.f};
        v8f acc1 = {0.f, 0.f, 0.f, 0.f, 0.f, 0.f, 0.f, 0.f};
        const int kb_off = half * 2;
        #pragma unroll 2
        for (int k0 = 0; k0 < RED; k0 += 8) {
            {
                const int kb = k0 + kb_off;
                v2f a, bb;
                a.x = sm_t[l16 * TSTR + kb];
                a.y = sm_t[l16 * TSTR + kb + 1];
                const float* wp = w2 + (col0 + l16) * RED + kb;
                bb.x = wp[0];
                bb.y = wp[1];
                acc0 = __builtin_amdgcn_wmma_f32_16x16x4_f32(
                    false, a, false, bb, (short)0, acc0, false, false);
            }
            {
                const int kb = k0 + 4 + kb_off;
                v2f a, bb;
                a.x = sm_t[l16 * TSTR + kb];
                a.y = sm_t[l16 * TSTR + kb + 1];
                const float* wp = w2 + (col0 + l16) * RED + kb;
                bb.x = wp[0];
                bb.y = wp[1];
                acc1 = __builtin_amdgcn_wmma_f32_16x16x4_f32(
                    false, a, false, bb, (short)0, acc1, false, false);
            }
        }
        const float bias = b2[col0 + l16];
        #pragma unroll
        for (int r = 0; r < 8; ++r) {
            const int m = r + half * 8;
            sm_wgt[m * KOUT + col0 + l16] = acc0[r] + acc1[r] + bias;
        }
    }
    __syncthreads();   // full 16x784 weight tile resident in LDS

    // ---- stage 3: involution: out[pix, g*16+cc] = sum_k wgt[pix, g*49+k] * x_patch ----
    for (int g = 0; g < NG; ++g) {
        // async-stage per-group x slab (16 ch x 7 rows x 20 cols), zero-padded borders
        for (int i = tid; i < GC * KW * SW; i += 128) {
            const int cc  = i / (KW * SW);
            const int rem = i - cc * (KW * SW);
            const int r   = rem / SW;
            const int cx  = rem - r * SW;
            const int hh  = h - 3 + r;
            const int ww  = w0 - 3 + cx;
            if (hh >= 0 && hh < HH && ww >= 0 && ww < WW) {
                const unsigned voff =
                    4u * (unsigned)(((b * CCH + g * GC + cc) * HH + hh) * WW + ww);
                async_g2l_b32((unsigned)(uintptr_t)(&sm_slab[i]), voff, x);
            } else {
                sm_slab[i] = 0.f;
            }
        }
        wait_async0();
        __syncthreads();

        for (int i = tid; i < TW * GC; i += 128) {       // 224 outputs / block
            const int pix = i % TW;
            const int cc  = i / TW;
            const float* wrow = &sm_wgt[pix * KOUT + g * KK2];
            const float* sl   = &sm_slab[cc * (KW * SW)];
            float acc = 0.f;
            #pragma unroll
            for (int kh = 0; kh < KW; ++kh)
                #pragma unroll
                for (int kw = 0; kw < KW; ++kw)
                    acc = fmaf(wrow[kh * KW + kw], sl[kh * SW + pix + kw], acc);
            out[((b * CCH + g * GC + cc) * HH + h) * WW + (w0 + pix)] = acc;
        }
        __syncthreads();   // protect slab before next group's load
    }
}

extern "C" void kernel_launch(void* const* d_in, const int* in_sizes, int n_in,
                              void* d_out, int out_size, void* d_ws, size_t ws_size,
                              hipStream_t stream) {
    const float* x  = (const float*)d_in[0];
    const float* w1 = (const float*)d_in[1];
    const float* b1 = (const float*)d_in[2];
    const float* w2 = (const float*)d_in[3];
    const float* b2 = (const float*)d_in[4];
    float* out = (float*)d_out;

    // 4 batches * 56 rows * 4 strips of 14 pixels = 896 blocks, 128 threads (4 wave32)
    dim3 grid(4 * HH * 4);
    dim3 block(128);
    hipLaunchKernelGGL(involution_fused_kernel, grid, block, 0, stream,
                       x, w1, b1, w2, b2, out);
}